// MultiHeadAttention_16690242913217
// MI455X (gfx1250) — compile-verified
//
#include <hip/hip_runtime.h>

// MI455X / gfx1250, wave32. f32 WMMA 16x16x4 per CDNA5 ISA 7.12.
typedef float v2f __attribute__((ext_vector_type(2)));
typedef float v8f __attribute__((ext_vector_type(8)));

__device__ __forceinline__ v8f wmma4(v2f a, v2f b, v8f c) {
  // (neg_a, A, neg_b, B, c_mod, C, reuse_a, reuse_b)
  return __builtin_amdgcn_wmma_f32_16x16x4_f32(false, a, false, b, (short)0, c,
                                               false, false);
}

// ---- CDNA5 async global->LDS (ASYNCcnt-tracked, bypasses VGPRs) ----
__device__ __forceinline__ unsigned ldsOff(const void* p) {
  // flat address of LDS = {aperture, offset}; low 32 bits are the LDS offset
  return (unsigned)(unsigned long long)p;
}
__device__ __forceinline__ void asyncLdB32(unsigned lds, unsigned goff,
                                           const void* base) {
  asm volatile("global_load_async_to_lds_b32 %0, %1, %2"
               :: "v"(lds), "v"(goff), "s"(base) : "memory");
}
__device__ __forceinline__ void asyncLdB128(unsigned lds, unsigned goff,
                                            const void* base) {
  asm volatile("global_load_async_to_lds_b128 %0, %1, %2"
               :: "v"(lds), "v"(goff), "s"(base) : "memory");
}
__device__ __forceinline__ void waitAsync0() {
  asm volatile("s_wait_asynccnt 0x0" ::: "memory");
}

#define L_SEQ 2048
#define D_MODEL 1024
#define NHEAD 16
#define DEPTH 64
#define SSTR 2052  // padded LDS stride for 2048-wide score rows (bank-safe)
#define KSTR 68    // padded LDS stride for 64-wide K/Q rows (17x16B aligned)
#define ASTR 17    // padded LDS stride for 16-wide A-tile rows
#define BPSTR 160  // K-paired B tile: pair-row stride (pair+1 -> +32 banks)

// ---------------------------------------------------------------------------
// C[M,N] = A[M,K] @ B[K,N] + bias[N]   (row-major f32)
// grid: (N/64, M/64); block: 128 = 4 waves; each wave computes 16x64 of C.
// Double-buffered async staging; K-paired B tile (one ds_load_2addr_b64 feeds
// two WMMAs). Fragments preloaded in bursts so WMMAs issue back-to-back.
// ---------------------------------------------------------------------------
__global__ __launch_bounds__(128) void gemm_bias(const float* __restrict__ A,
                                                 const float* __restrict__ B,
                                                 const float* __restrict__ bias,
                                                 float* __restrict__ C,
                                                 int M, int N, int K) {
  __shared__ float As[2][64 * ASTR];
  __shared__ float Bs[2][8 * BPSTR];

  const int tid = threadIdx.x;
  const int wave = tid >> 5;
  const int lane = tid & 31;
  const int ln15 = lane & 15;
  const int khalf = (lane >> 4) << 1;  // 0 | 2 : fragment K split across halves
  const int m0 = blockIdx.y * 64;
  const int n0 = blockIdx.x * 64;

  auto stage = [&](int bb, int k0) {
    for (int i = tid; i < 64 * 16; i += 128) {
      int r = i >> 4, c = i & 15;
      asyncLdB32(ldsOff(&As[bb][r * ASTR + c]),
                 (unsigned)(((size_t)(m0 + r) * K + k0 + c) * 4), A);
    }
    for (int i = tid; i < 16 * 64; i += 128) {
      int r = i >> 6, c = i & 63;  // K-paired: [pair][col*2 + (k&1)]
      asyncLdB32(ldsOff(&Bs[bb][(r >> 1) * BPSTR + c * 2 + (r & 1)]),
                 (unsigned)(((size_t)(k0 + r) * N + n0 + c) * 4), B);
    }
  };

  v8f acc[4] = {};
  stage(0, 0);
  waitAsync0();
  __syncthreads();

  for (int k0 = 0; k0 < K; k0 += 16) {
    const int cur = (k0 >> 4) & 1;
    if (k0 + 16 < K) stage(cur ^ 1, k0 + 16);  // overlap DMA with WMMA

    // Burst-load all fragments for this 16-deep K slab, then burst WMMAs.
    v2f afr[4];
    v2f bfr[4][4];
#pragma unroll
    for (int kq = 0; kq < 4; kq++) {
      const int kk = kq * 4;
      afr[kq].x = As[cur][(wave * 16 + ln15) * ASTR + kk + khalf];
      afr[kq].y = As[cur][(wave * 16 + ln15) * ASTR + kk + khalf + 1];
      const int bp = (kk + khalf) >> 1;
#pragma unroll
      for (int nt = 0; nt < 4; nt++)
        bfr[kq][nt] = *(const v2f*)&Bs[cur][bp * BPSTR + (nt * 16 + ln15) * 2];
    }
#pragma unroll
    for (int kq = 0; kq < 4; kq++)
#pragma unroll
      for (int nt = 0; nt < 4; nt++)
        acc[nt] = wmma4(afr[kq], bfr[kq][nt], acc[nt]);

    waitAsync0();    // our asm async ops are invisible to the compiler
    __syncthreads();
  }

  const int rowOff = (lane >> 4) * 8;
#pragma unroll
  for (int nt = 0; nt < 4; nt++) {
#pragma unroll
    for (int rr = 0; rr < 8; rr++) {
      int m = m0 + wave * 16 + rowOff + rr;
      int n = n0 + nt * 16 + ln15;
      C[(size_t)m * N + n] = acc[nt][rr] + bias[n];
    }
  }
}

// ---------------------------------------------------------------------------
// Fused attention for one (batch, head, 16-query-row panel):
//   phase 1: Ss = Qh @ Kh^T * 0.125 + mask*-1e9  (WMMA, dbl-buffered async K)
//   phase 2: row softmax in LDS; weights hit HBM exactly once
//   phase 3: attn = P @ Vh                       (WMMA, dbl-buffered async V)
// 16x2048 f32 score panel lives entirely in CDNA5's 320KB LDS (~131KB).
// grid: (L/16, B*H); block: 128 = 4 waves.
// ---------------------------------------------------------------------------
__global__ __launch_bounds__(128) void fused_attn(const float* __restrict__ Qp,
                                                  const float* __restrict__ Kp,
                                                  const float* __restrict__ Vp,
                                                  const int* __restrict__ mask,
                                                  float* __restrict__ S,
                                                  float* __restrict__ Ab) {
  __shared__ float Ss[16 * SSTR];        // score panel, ~131KB
  __shared__ float KVs[2][64 * KSTR];    // phase1: K chunks; phase3: V buffers
  __shared__ float Qs[16 * KSTR];
  __shared__ float red[128];

  const int tid = threadIdx.x;
  const int wave = tid >> 5;
  const int lane = tid & 31;
  const int ln15 = lane & 15;
  const int khalf = (lane >> 4) << 1;
  const int rowOff = (lane >> 4) * 8;

  const int bh = blockIdx.y;
  const int b = bh >> 4;
  const int h = bh & 15;
  const int q0 = blockIdx.x * 16;

  const size_t headBase = (size_t)b * L_SEQ * D_MODEL + (size_t)h * DEPTH;
  const size_t maskBase = (size_t)b * L_SEQ * L_SEQ;

  // Q panel: 256 x 16B async chunks (KSTR rows are 17*16B -> aligned).
  for (int ch = tid; ch < 16 * 16; ch += 128) {
    int r = ch >> 4, c4 = (ch & 15) * 4;
    asyncLdB128(ldsOff(&Qs[r * KSTR + c4]),
                (unsigned)((headBase + (size_t)(q0 + r) * D_MODEL + c4) * 4), Qp);
  }

  // ---- phase 1: scores, 64-key chunks, double buffered ----
  auto stageK = [&](int bb, int c0) {
    for (int ch = tid; ch < 64 * 16; ch += 128) {
      int r = ch >> 4, c4 = (ch & 15) * 4;
      asyncLdB128(ldsOff(&KVs[bb][r * KSTR + c4]),
                  (unsigned)((headBase + (size_t)(c0 + r) * D_MODEL + c4) * 4), Kp);
    }
  };
  stageK(0, 0);
  waitAsync0();
  __syncthreads();

  for (int c0 = 0; c0 < L_SEQ; c0 += 64) {
    const int cur = (c0 >> 6) & 1;
    if (c0 + 64 < L_SEQ) stageK(cur ^ 1, c0 + 64);

    v8f acc = {};
#pragma unroll
    for (int dq = 0; dq < 4; dq++) {
      // Burst-load 4 fragment pairs, then 4 back-to-back WMMAs.
      v2f a4[4], b4[4];
#pragma unroll
      for (int j = 0; j < 4; j++) {
        const int d0 = dq * 16 + j * 4;
        a4[j].x = Qs[ln15 * KSTR + d0 + khalf];
        a4[j].y = Qs[ln15 * KSTR + d0 + khalf + 1];
        b4[j].x = KVs[cur][(wave * 16 + ln15) * KSTR + d0 + khalf];
        b4[j].y = KVs[cur][(wave * 16 + ln15) * KSTR + d0 + khalf + 1];
      }
#pragma unroll
      for (int j = 0; j < 4; j++) acc = wmma4(a4[j], b4[j], acc);
    }

#pragma unroll
    for (int rr = 0; rr < 8; rr++) {
      int qr = rowOff + rr;
      int key = c0 + wave * 16 + ln15;
      float sv = acc[rr] * 0.125f +
                 (float)mask[maskBase + (size_t)(q0 + qr) * L_SEQ + key] * (-1.0e9f);
      Ss[qr * SSTR + key] = sv;
    }
    waitAsync0();
    __syncthreads();
  }

  // ---- phase 2: softmax (16 rows x 2048, 8 threads per row) ----
  {
    const int r = tid >> 3;
    const int subl = tid & 7;
    float m = -3.0e38f;
    for (int i = subl; i < L_SEQ; i += 8) m = fmaxf(m, Ss[r * SSTR + i]);
    red[tid] = m;
    __syncthreads();
    if (subl == 0) {
      float mm = red[tid];
#pragma unroll
      for (int j = 1; j < 8; j++) mm = fmaxf(mm, red[tid + j]);
      red[tid] = mm;
    }
    __syncthreads();
    m = red[r * 8];
    float sum = 0.0f;
    for (int i = subl; i < L_SEQ; i += 8) {
      float e = __expf(Ss[r * SSTR + i] - m);
      Ss[r * SSTR + i] = e;
      sum += e;
    }
    __syncthreads();
    red[tid] = sum;
    __syncthreads();
    if (subl == 0) {
      float s = 0.0f;
#pragma unroll
      for (int j = 0; j < 8; j++) s += red[tid + j];
      red[tid] = 1.0f / s;
    }
    __syncthreads();
    // Normalize in LDS and stream the weights to HBM once (coalesced).
    for (int idx = tid; idx < 16 * L_SEQ; idx += 128) {
      int rr = idx >> 11, i = idx & (L_SEQ - 1);
      float p = Ss[rr * SSTR + i] * red[rr * 8];
      Ss[rr * SSTR + i] = p;
      S[((size_t)bh * L_SEQ + (q0 + rr)) * L_SEQ + i] = p;
    }
  }
  __syncthreads();

  // ---- phase 3: attn = P @ V, K-paired V chunks, double buffered ----
  float* Vb0 = &KVs[0][0];
  float* Vb1 = &KVs[0][0] + 1440;  // 5760B offset: keeps 8B/16B alignment
  auto stageV = [&](float* dst, int k0) {
    for (int i = tid; i < 16 * 64; i += 128) {
      int r = i >> 6, c = i & 63;  // K-paired layout for single-b64 fragments
      asyncLdB32(ldsOff(&dst[(r >> 1) * BPSTR + c * 2 + (r & 1)]),
                 (unsigned)((headBase + (size_t)(k0 + r) * D_MODEL + c) * 4), Vp);
    }
  };

  v8f acc = {};
  stageV(Vb0, 0);
  waitAsync0();
  __syncthreads();

  for (int k0 = 0; k0 < L_SEQ; k0 += 16) {
    float* curV = ((k0 >> 4) & 1) ? Vb1 : Vb0;
    float* nxtV = ((k0 >> 4) & 1) ? Vb0 : Vb1;
    if (k0 + 16 < L_SEQ) stageV(nxtV, k0 + 16);

    // Burst-load all fragments for this 16-deep slab, then burst WMMAs.
    v2f a4[4], b4[4];
#pragma unroll
    for (int j = 0; j < 4; j++) {
      const int kk = j * 4;
      a4[j].x = Ss[ln15 * SSTR + k0 + kk + khalf];
      a4[j].y = Ss[ln15 * SSTR + k0 + kk + khalf + 1];
      b4[j] = *(const v2f*)&curV[((kk + khalf) >> 1) * BPSTR +
                                 (wave * 16 + ln15) * 2];
    }
#pragma unroll
    for (int j = 0; j < 4; j++) acc = wmma4(a4[j], b4[j], acc);

    waitAsync0();
    __syncthreads();
  }
#pragma unroll
  for (int rr = 0; rr < 8; rr++) {
    Ab[headBase + (size_t)(q0 + rowOff + rr) * D_MODEL + wave * 16 + ln15] =
        acc[rr];
  }
}

// ---------------------------------------------------------------------------
extern "C" void kernel_launch(void* const* d_in, const int* in_sizes, int n_in,
                              void* d_out, int out_size, void* d_ws, size_t ws_size,
                              hipStream_t stream) {
  (void)in_sizes; (void)n_in; (void)out_size; (void)ws_size;
  const float* q   = (const float*)d_in[0];
  const float* k   = (const float*)d_in[1];
  const float* v   = (const float*)d_in[2];
  const int*   msk = (const int*)  d_in[3];
  const float* Wq  = (const float*)d_in[4];
  const float* bq  = (const float*)d_in[5];
  const float* Wk  = (const float*)d_in[6];
  const float* bk  = (const float*)d_in[7];
  const float* Wv  = (const float*)d_in[8];
  const float* bv  = (const float*)d_in[9];
  const float* Wo  = (const float*)d_in[10];
  const float* bo  = (const float*)d_in[11];

  const int Btch = 2, Dm = D_MODEL, NT = Btch * L_SEQ;  // 4096 tokens

  float* out = (float*)d_out;
  float* S   = out + (size_t)NT * Dm;  // attn_weights section of tuple output

  float* ws = (float*)d_ws;
  float* Qp = ws;                    // (NT, D) projected Q
  float* Kp = Qp + (size_t)NT * Dm;  // (NT, D) projected K
  float* Vp = Kp + (size_t)NT * Dm;  // (NT, D) projected V
  float* Ab = Vp + (size_t)NT * Dm;  // (NT, D) concat-head attention

  dim3 gG(Dm / 64, NT / 64);
  gemm_bias<<<gG, 128, 0, stream>>>(q, Wq, bq, Qp, NT, Dm, Dm);
  gemm_bias<<<gG, 128, 0, stream>>>(k, Wk, bk, Kp, NT, Dm, Dm);
  gemm_bias<<<gG, 128, 0, stream>>>(v, Wv, bv, Vp, NT, Dm, Dm);

  dim3 gA(L_SEQ / 16, Btch * NHEAD);
  fused_attn<<<gA, 128, 0, stream>>>(Qp, Kp, Vp, msk, S, Ab);

  gemm_bias<<<gG, 128, 0, stream>>>(Ab, Wo, bo, out, NT, Dm, Dm);
}